// TrigoLinear_37520834298117
// MI455X (gfx1250) — compile-verified
//
#include <hip/hip_runtime.h>
#include <hip/hip_bf16.h>

#define B_TOT 2048
#define I_TOT 256
#define O_TOT 512

#define BT 128          // batch tile per block
#define OT 32           // output tile per block
#define KC 32           // i-chunk
#define NCHUNK (I_TOT / KC)

#define XPAD 36         // row stride (floats) for s_x: multiple of 4 -> 16B-aligned b128 slots
#define WPAD 33         // row stride for s_w (float2) and s_b (float): odd -> conflict-free reads

// ---- CDNA5 async global->LDS path (ASYNCcnt), with safe fallback --------------------
#if defined(__has_builtin)
#if __has_builtin(__builtin_amdgcn_global_load_async_to_lds_b32) && \
    __has_builtin(__builtin_amdgcn_global_load_async_to_lds_b64) && \
    __has_builtin(__builtin_amdgcn_global_load_async_to_lds_b128) && \
    __has_builtin(__builtin_amdgcn_s_wait_asynccnt)
#define USE_ASYNC_LDS 1
#endif
#endif

typedef int v2i __attribute__((ext_vector_type(2)));
typedef int v4i __attribute__((ext_vector_type(4)));

typedef __attribute__((address_space(1))) int  gi32_t;   // global, typed per builtin signature
typedef __attribute__((address_space(1))) v2i  gi64_t;
typedef __attribute__((address_space(1))) v4i  gi128_t;
typedef __attribute__((address_space(3))) int  li32_t;   // LDS, typed (converts to AS3 void* if needed)
typedef __attribute__((address_space(3))) v2i  li64_t;
typedef __attribute__((address_space(3))) v4i  li128_t;

__device__ __forceinline__ void copy_b32(const float* g, float* l) {
#ifdef USE_ASYNC_LDS
  __builtin_amdgcn_global_load_async_to_lds_b32((gi32_t*)g, (li32_t*)l, 0, 0);
#else
  *l = *g;
#endif
}
__device__ __forceinline__ void copy_b64(const float2* g, float2* l) {
#ifdef USE_ASYNC_LDS
  __builtin_amdgcn_global_load_async_to_lds_b64((gi64_t*)g, (li64_t*)l, 0, 0);
#else
  *l = *g;
#endif
}
__device__ __forceinline__ void copy_b128(const float4* g, float4* l) {
#ifdef USE_ASYNC_LDS
  __builtin_amdgcn_global_load_async_to_lds_b128((gi128_t*)g, (li128_t*)l, 0, 0);
#else
  *l = *g;
#endif
}
__device__ __forceinline__ void wait_async_lds() {
#ifdef USE_ASYNC_LDS
  __builtin_amdgcn_s_wait_asynccnt(0);
#endif
}

// -------------------------------------------------------------------------------------
__global__ __launch_bounds__(256, 4)
void TrigoLinear_kernel(const float* __restrict__ x,
                        const float* __restrict__ weight,
                        const float* __restrict__ bias,
                        float* __restrict__ out) {
  __shared__ float  s_x[2][BT][XPAD];   // x[b][i] tile
  __shared__ float2 s_w[2][OT][WPAD];   // (w_out, w_sin) per (o,i)
  __shared__ float  s_b[2][OT][WPAD];   // b_sin per (o,i)

  const int tid = threadIdx.x;
  const int tx  = tid & 15;   // o-group: thread covers o = o0 + tx*2 + {0,1}
  const int ty  = tid >> 4;   // b-group: thread covers b = b0 + ty*8 + {0..7}
  const int o0  = blockIdx.x * OT;
  const int b0  = blockIdx.y * BT;

  const float2* wpair = (const float2*)weight;   // [O][I] of (w_out, w_sin)

  auto stage = [&](int c, int buf) {
    const int ic0 = c * KC;
    // x tile: BT x KC floats = 1024 float4, 4 per thread; lanes walk contiguous global bytes
#pragma unroll
    for (int r = 0; r < 4; ++r) {
      int e  = r * 256 + tid;
      int bb = e >> 3;        // 0..127
      int kq = e & 7;         // float4 index within chunk
      copy_b128((const float4*)(x + (size_t)(b0 + bb) * I_TOT + ic0 + kq * 4),
                (float4*)&s_x[buf][bb][kq * 4]);
    }
    // weight pairs + sin-bias: OT x KC = 1024 each, 4 per thread
#pragma unroll
    for (int r = 0; r < 4; ++r) {
      int e  = r * 256 + tid;
      int oo = e >> 5;        // 0..31
      int kk = e & 31;        // 0..31
      copy_b64(wpair + (size_t)(o0 + oo) * I_TOT + ic0 + kk, &s_w[buf][oo][kk]);
      copy_b32(bias  + (size_t)(o0 + oo) * (I_TOT + 1) + ic0 + kk, &s_b[buf][oo][kk]);
    }
  };

  float acc[2][8];
#pragma unroll
  for (int oo = 0; oo < 2; ++oo)
#pragma unroll
    for (int j = 0; j < 8; ++j) acc[oo][j] = 0.0f;

  stage(0, 0);
  wait_async_lds();
  __syncthreads();

#pragma unroll 1
  for (int c = 0; c < NCHUNK; ++c) {
    const int buf = c & 1;
    if (c + 1 < NCHUNK) stage(c + 1, buf ^ 1);   // async prefetch next chunk

#pragma unroll 4
    for (int kk = 0; kk < KC; ++kk) {
      float xv[8];
#pragma unroll
      for (int j = 0; j < 8; ++j) xv[j] = s_x[buf][ty * 8 + j][kk];  // broadcast reads
#pragma unroll
      for (int oo = 0; oo < 2; ++oo) {
        float2 w  = s_w[buf][tx * 2 + oo][kk];   // conflict-free (odd stride)
        float  bs = s_b[buf][tx * 2 + oo][kk];
#pragma unroll
        for (int j = 0; j < 8; ++j) {
          float s = __sinf(__builtin_fmaf(xv[j], w.y, bs));  // v_sin_f32 path
          acc[oo][j] = __builtin_fmaf(w.x, s, acc[oo][j]);
        }
      }
    }

    wait_async_lds();     // drain this wave's async stores into LDS
    __syncthreads();      // then everyone may read the freshly staged buffer
  }

  // epilogue: add b_out, write coalesced float2 (two consecutive o per thread)
  const float bo0 = bias[(size_t)(o0 + tx * 2 + 0) * (I_TOT + 1) + I_TOT];
  const float bo1 = bias[(size_t)(o0 + tx * 2 + 1) * (I_TOT + 1) + I_TOT];
#pragma unroll
  for (int j = 0; j < 8; ++j) {
    const int b = b0 + ty * 8 + j;
    float2 v;
    v.x = acc[0][j] + bo0;
    v.y = acc[1][j] + bo1;
    *(float2*)&out[(size_t)b * O_TOT + o0 + tx * 2] = v;
  }
}

// -------------------------------------------------------------------------------------
extern "C" void kernel_launch(void* const* d_in, const int* in_sizes, int n_in,
                              void* d_out, int out_size, void* d_ws, size_t ws_size,
                              hipStream_t stream) {
  (void)in_sizes; (void)n_in; (void)d_ws; (void)ws_size; (void)out_size;
  const float* x      = (const float*)d_in[0];
  const float* weight = (const float*)d_in[1];
  const float* bias   = (const float*)d_in[2];
  float* out          = (float*)d_out;

  dim3 grid(O_TOT / OT, B_TOT / BT);   // (16, 16) = 256 blocks
  TrigoLinear_kernel<<<grid, 256, 0, stream>>>(x, weight, bias, out);
}